// PytorchMambaOneStepSSMModel_60224031424806
// MI455X (gfx1250) — compile-verified
//
#include <hip/hip_runtime.h>

// Problem constants (from reference)
#define B_      2
#define LSEQ    2048
#define DMODEL  2560
#define DIN     (2*DMODEL)      // 5120
#define NST     16
#define DTR     160             // DT_RANK = 2560/16
#define EPROJ   (DTR + 2*NST)   // 192
#define MTOT    (B_*LSEQ)       // 4096 rows for the GEMMs
#define GCH     16              // scan chunks
#define CHLEN   (LSEQ/GCH)      // 128 steps per chunk

typedef float v2f __attribute__((ext_vector_type(2)));
typedef float v8f __attribute__((ext_vector_type(8)));
typedef int   v4i __attribute__((ext_vector_type(4)));

#if defined(__has_builtin)
#  if __has_builtin(__builtin_amdgcn_global_load_async_to_lds_b128)
#    define HAVE_ASYNC_LDS 1
#  endif
#endif
#ifndef HAVE_ASYNC_LDS
#  define HAVE_ASYNC_LDS 0
#endif

#if HAVE_ASYNC_LDS
typedef __attribute__((address_space(1))) v4i* g4p;   // global (AS1) int4*
typedef __attribute__((address_space(3))) v4i* l4p;   // LDS (AS3) int4*
__device__ __forceinline__ void async_copy16(const float* gsrc, float4* ldst) {
    __builtin_amdgcn_global_load_async_to_lds_b128((g4p)gsrc, (l4p)ldst, 0, 0);
}
#endif

__device__ __forceinline__ v2f ld2(const float* p) {
    return *reinterpret_cast<const v2f*>(p);
}

// ---------------------------------------------------------------------------
// Kernel 1: x_dbl[m, e] = sum_k x[m,k] * W1[e,k]   (NT GEMM, K=5120 contiguous)
// One wave computes TWO 16x16 tiles (m0, m0+16) sharing the B (weight) fetch:
// per K-step 3x b64 loads feed 2x V_WMMA_F32_16X16X4_F32.
// A layout (16x4 f32): lane&15 = row, (lane>>4)*2 = K pair -> float2 load.
// ---------------------------------------------------------------------------
__global__ __launch_bounds__(256) void xproj_gemm(const float* __restrict__ X,
                                                  const float* __restrict__ W1,
                                                  float* __restrict__ XD) {
    const int lane = threadIdx.x & 31;
    const int wave = threadIdx.x >> 5;
    const int tile = blockIdx.x * 8 + wave;     // 128*12 = 1536 wave-tiles
    const int mt = tile / (EPROJ / 16);         // 0..127 (32-row blocks)
    const int et = tile % (EPROJ / 16);         // 0..11
    const int m0 = mt * 32, e0 = et * 16;
    const int row = lane & 15;
    const int kh  = (lane >> 4) * 2;

    const float* pa0 = X  + (size_t)(m0 + row) * DIN + kh;
    const float* pa1 = pa0 + (size_t)16 * DIN;
    const float* pb  = W1 + (size_t)(e0 + row) * DIN + kh;

    v8f c0 = {0.f, 0.f, 0.f, 0.f, 0.f, 0.f, 0.f, 0.f};
    v8f c1 = {0.f, 0.f, 0.f, 0.f, 0.f, 0.f, 0.f, 0.f};
#pragma unroll 8
    for (int k = 0; k < DIN; k += 4) {
        v2f a0 = ld2(pa0);
        v2f a1 = ld2(pa1);
        v2f b  = ld2(pb);
        c0 = __builtin_amdgcn_wmma_f32_16x16x4_f32(false, a0, false, b,
                                                   (short)0, c0, false, false);
        c1 = __builtin_amdgcn_wmma_f32_16x16x4_f32(false, a1, false, b,
                                                   (short)0, c1, false, false);
        pa0 += 4; pa1 += 4; pb += 4;
    }
    // C/D layout: VGPR i -> M = i (lanes 0-15) or 8 + i (lanes 16-31)
    const int mrow = m0 + (lane >> 4) * 8;
#pragma unroll
    for (int i = 0; i < 8; ++i) {
        XD[(size_t)(mrow + i) * EPROJ + e0 + row]      = c0[i];
        XD[(size_t)(mrow + 16 + i) * EPROJ + e0 + row] = c1[i];
    }
}

// ---------------------------------------------------------------------------
// Kernel 2: delta[m, d] = softplus( sum_r x_dbl[m, r] * W2[d, r] + bias[d] )
// K = 160, same 2xM-blocked NT-WMMA pattern; fused bias + softplus epilogue.
// ---------------------------------------------------------------------------
__global__ __launch_bounds__(256) void delta_gemm(const float* __restrict__ XD,
                                                  const float* __restrict__ W2,
                                                  const float* __restrict__ bias,
                                                  float* __restrict__ DELTA) {
    const int lane = threadIdx.x & 31;
    const int wave = threadIdx.x >> 5;
    const int tile = blockIdx.x * 8 + wave;     // 128*320 = 40960 wave-tiles
    const int mt = tile / (DIN / 16);           // 0..127
    const int dt = tile % (DIN / 16);           // 0..319
    const int m0 = mt * 32, d0 = dt * 16;
    const int row = lane & 15;
    const int kh  = (lane >> 4) * 2;

    const float* pa0 = XD + (size_t)(m0 + row) * EPROJ + kh;  // dt_raw cols [0,160)
    const float* pa1 = pa0 + (size_t)16 * EPROJ;
    const float* pb  = W2 + (size_t)(d0 + row) * DTR + kh;

    v8f c0 = {0.f, 0.f, 0.f, 0.f, 0.f, 0.f, 0.f, 0.f};
    v8f c1 = {0.f, 0.f, 0.f, 0.f, 0.f, 0.f, 0.f, 0.f};
#pragma unroll
    for (int k = 0; k < DTR; k += 4) {
        v2f a0 = ld2(pa0);
        v2f a1 = ld2(pa1);
        v2f b  = ld2(pb);
        c0 = __builtin_amdgcn_wmma_f32_16x16x4_f32(false, a0, false, b,
                                                   (short)0, c0, false, false);
        c1 = __builtin_amdgcn_wmma_f32_16x16x4_f32(false, a1, false, b,
                                                   (short)0, c1, false, false);
        pa0 += 4; pa1 += 4; pb += 4;
    }
    const int mrow = m0 + (lane >> 4) * 8;
    const int d = d0 + row;
    const float bv = bias[d];
#pragma unroll
    for (int i = 0; i < 8; ++i) {
        float v0 = c0[i] + bv;
        float v1 = c1[i] + bv;
        float s0 = (v0 > 20.f) ? v0 : ((v0 < -20.f) ? __expf(v0) : __logf(1.f + __expf(v0)));
        float s1 = (v1 > 20.f) ? v1 : ((v1 < -20.f) ? __expf(v1) : __logf(1.f + __expf(v1)));
        DELTA[(size_t)(mrow + i) * DIN + d]      = s0;
        DELTA[(size_t)(mrow + 16 + i) * DIN + d] = s1;
    }
}

// ---------------------------------------------------------------------------
// Scan pass 1: per (b, chunk g, d) compute, starting from h=0:
//   chunk transition  ap[n] = exp2(aa[n] * sum_l dt_l)   (exponent sum)
//   local end state   h[n]
// lane -> d (coalesced x/delta); B_t staged in LDS (async if available),
// read back as ds_load_b128 broadcasts.
// ---------------------------------------------------------------------------
__global__ __launch_bounds__(256) void scan_pass1(const float* __restrict__ X,
                                                  const float* __restrict__ XD,
                                                  const float* __restrict__ DELTA,
                                                  const float* __restrict__ A_log,
                                                  float* __restrict__ CA,
                                                  float* __restrict__ CHS) {
    __shared__ float4 sB[CHLEN][NST / 4];       // 8 KB, row = 16 contiguous floats
    const int tid  = threadIdx.x;
    const int dblk = blockIdx.x % (DIN / 256);
    const int g    = (blockIdx.x / (DIN / 256)) % GCH;
    const int b    =  blockIdx.x / ((DIN / 256) * GCH);
    const int d  = dblk * 256 + tid;
    const int l0 = g * CHLEN;

#if HAVE_ASYNC_LDS
    for (int idx = tid; idx < CHLEN * (NST / 4); idx += 256) {
        int l = idx >> 2, q = idx & 3;
        async_copy16(&XD[(size_t)(b * LSEQ + l0 + l) * EPROJ + DTR + q * 4],
                     &sB[l][q]);
    }
    asm volatile("s_wait_asynccnt 0x0" ::: "memory");
#else
    for (int idx = tid; idx < CHLEN * NST; idx += 256) {
        int l = idx >> 4, n = idx & 15;
        (&sB[0][0].x)[idx] = XD[(size_t)(b * LSEQ + l0 + l) * EPROJ + DTR + n];
    }
#endif
    __syncthreads();

    float aa[NST];
#pragma unroll
    for (int n = 0; n < NST; ++n)
        aa[n] = -__expf(A_log[(size_t)d * NST + n]) * 1.44269504088896f; // A*log2(e)

    float h[NST];
#pragma unroll
    for (int n = 0; n < NST; ++n) h[n] = 0.f;
    float sdt = 0.f;

    const size_t base = (size_t)(b * LSEQ + l0) * DIN + d;
    for (int l = 0; l < CHLEN; ++l) {
        const size_t off = base + (size_t)l * DIN;
        if (l + 8 < CHLEN) {
            __builtin_prefetch(&DELTA[off + 8 * (size_t)DIN], 0, 0);
            __builtin_prefetch(&X[off + 8 * (size_t)DIN], 0, 0);
        }
        float dtv = DELTA[off];
        float xt  = X[off];
        float u   = dtv * xt;
        sdt += dtv;
#pragma unroll
        for (int q = 0; q < NST / 4; ++q) {
            float4 bb = sB[l][q];                  // ds_load_b128 broadcast
            const float* bp = reinterpret_cast<const float*>(&bb);
#pragma unroll
            for (int c = 0; c < 4; ++c) {
                const int n = q * 4 + c;
                float dA = exp2f(dtv * aa[n]);
                h[n] = dA * h[n] + u * bp[c];
            }
        }
    }
    const size_t so = (size_t)(b * GCH + g) * DIN + d;
#pragma unroll
    for (int n = 0; n < NST; ++n) {
        CA [so + (size_t)n * (B_ * GCH) * DIN] = exp2f(sdt * aa[n]);
        CHS[so + (size_t)n * (B_ * GCH) * DIN] = h[n];
    }
}

// ---------------------------------------------------------------------------
// Scan pass 2: fold chunk prefixes (j < g) into h0, replay the chunk with the
// correct initial state, emit y_t = C_t . h_t + x_t * D[d].
// sBC row = the 32 contiguous floats x_dbl[m, 160:192] = [B_t | C_t].
// ---------------------------------------------------------------------------
__global__ __launch_bounds__(256) void scan_pass2(const float* __restrict__ X,
                                                  const float* __restrict__ XD,
                                                  const float* __restrict__ DELTA,
                                                  const float* __restrict__ A_log,
                                                  const float* __restrict__ Dvec,
                                                  const float* __restrict__ CA,
                                                  const float* __restrict__ CHS,
                                                  float* __restrict__ OUT) {
    __shared__ float4 sBC[CHLEN][2 * NST / 4];  // 16 KB
    const int tid  = threadIdx.x;
    const int dblk = blockIdx.x % (DIN / 256);
    const int g    = (blockIdx.x / (DIN / 256)) % GCH;
    const int b    =  blockIdx.x / ((DIN / 256) * GCH);
    const int d  = dblk * 256 + tid;
    const int l0 = g * CHLEN;

#if HAVE_ASYNC_LDS
    for (int idx = tid; idx < CHLEN * 8; idx += 256) {
        int l = idx >> 3, q = idx & 7;
        async_copy16(&XD[(size_t)(b * LSEQ + l0 + l) * EPROJ + DTR + q * 4],
                     &sBC[l][q]);
    }
    asm volatile("s_wait_asynccnt 0x0" ::: "memory");
#else
    for (int idx = tid; idx < CHLEN * 2 * NST; idx += 256) {
        int l = idx >> 5, j = idx & 31;
        (&sBC[0][0].x)[idx] = XD[(size_t)(b * LSEQ + l0 + l) * EPROJ + DTR + j];
    }
#endif
    __syncthreads();

    float aa[NST];
#pragma unroll
    for (int n = 0; n < NST; ++n)
        aa[n] = -__expf(A_log[(size_t)d * NST + n]) * 1.44269504088896f;

    // prefix-combine chunk states: h0 = Ac_j * h0 + Hc_j for j = 0..g-1
    float h[NST];
#pragma unroll
    for (int n = 0; n < NST; ++n) h[n] = 0.f;
    for (int j = 0; j < g; ++j) {
        const size_t so = (size_t)(b * GCH + j) * DIN + d;
#pragma unroll
        for (int n = 0; n < NST; ++n) {
            const size_t o = so + (size_t)n * (B_ * GCH) * DIN;
            h[n] = CA[o] * h[n] + CHS[o];
        }
    }

    const float Dd = Dvec[d];
    const size_t base = (size_t)(b * LSEQ + l0) * DIN + d;
    for (int l = 0; l < CHLEN; ++l) {
        const size_t off = base + (size_t)l * DIN;
        if (l + 8 < CHLEN) {
            __builtin_prefetch(&DELTA[off + 8 * (size_t)DIN], 0, 0);
            __builtin_prefetch(&X[off + 8 * (size_t)DIN], 0, 0);
        }
        float dtv = DELTA[off];
        float xt  = X[off];
        float u   = dtv * xt;
        float y   = 0.f;
#pragma unroll
        for (int q = 0; q < NST / 4; ++q) {
            float4 bb = sBC[l][q];                 // B_t quad
            float4 cc = sBC[l][4 + q];             // C_t quad
            const float* bp = reinterpret_cast<const float*>(&bb);
            const float* cp = reinterpret_cast<const float*>(&cc);
#pragma unroll
            for (int c = 0; c < 4; ++c) {
                const int n = q * 4 + c;
                float dA = exp2f(dtv * aa[n]);
                h[n] = dA * h[n] + u * bp[c];
                y   += h[n] * cp[c];
            }
        }
        OUT[off] = y + xt * Dd;
    }
}

// ---------------------------------------------------------------------------
// Launch pipeline (same stream => ordered): gemm1 -> gemm2 -> pass1 -> pass2
// Workspace (floats): x_dbl 786432 | delta 20971520 | CA 2621440 | CHS 2621440
// ---------------------------------------------------------------------------
extern "C" void kernel_launch(void* const* d_in, const int* in_sizes, int n_in,
                              void* d_out, int out_size, void* d_ws, size_t ws_size,
                              hipStream_t stream) {
    const float* x      = (const float*)d_in[0];   // (2, 2048, 5120)
    const float* A_log  = (const float*)d_in[1];   // (5120, 16)
    const float* Dvec   = (const float*)d_in[2];   // (5120,)
    const float* w1     = (const float*)d_in[3];   // (192, 5120)
    const float* w2     = (const float*)d_in[4];   // (5120, 160)
    const float* b2     = (const float*)d_in[5];   // (5120,)
    float* out = (float*)d_out;

    float* ws    = (float*)d_ws;
    float* xdbl  = ws;                                   // MTOT * EPROJ
    float* delta = xdbl  + (size_t)MTOT * EPROJ;         // MTOT * DIN
    float* cA    = delta + (size_t)MTOT * DIN;           // NST * B_*GCH * DIN
    float* cH    = cA    + (size_t)NST * B_ * GCH * DIN; // NST * B_*GCH * DIN

    // GEMM 1: 1536 wave-tiles (2 M-tiles each), 8 waves per block
    xproj_gemm<<<(MTOT / 32) * (EPROJ / 16) / 8, 256, 0, stream>>>(x, w1, xdbl);
    // GEMM 2: 40960 wave-tiles
    delta_gemm<<<(MTOT / 32) * (DIN / 16) / 8, 256, 0, stream>>>(xdbl, w2, b2, delta);
    // Chunked scan: 2 * 16 * 20 = 640 blocks each pass
    dim3 gs(B_ * GCH * (DIN / 256));
    scan_pass1<<<gs, 256, 0, stream>>>(x, xdbl, delta, A_log, cA, cH);
    scan_pass2<<<gs, 256, 0, stream>>>(x, xdbl, delta, A_log, Dvec, cA, cH, out);
}